// TemporalGNN_SoftSupervision_34866544509673
// MI455X (gfx1250) — compile-verified
//
#include <hip/hip_runtime.h>

// Problem dims (compile-time constants from the reference)
#define IN_DIM 128
#define HID    256
#define OUTD   128
#define BB     4
#define TT     512
#define NCLS   50
#define NPAD   52          // NCLS padded to multiple of 4 for K-loop
#define ROWS   (BB*TT)     // 2048
#define LN_EPS 1e-5f
#define ADJ_SCALE 0.0625f  // HID^-0.5 = 1/16

typedef float v2f __attribute__((ext_vector_type(2)));
typedef float v8f __attribute__((ext_vector_type(8)));

// ---------------------------------------------------------------------------
// fp32 WMMA helpers: one wave computes a 16x16 fp32 tile via
// V_WMMA_F32_16X16X4_F32, stepping K by 4.
// A-frag layout (ISA 7.12.2, 32-bit 16x4): lanes 0-15 -> K=0,1 ; lanes 16-31 -> K=2,3
// B-frag: row K striped across 16 lanes per VGPR, mirrored halves.
// C/D: VGPR i holds M = i + 8*(lane>=16), N = lane&15.
// ---------------------------------------------------------------------------
template <typename LA, typename LB>
__device__ inline v8f wmma_loop(LA loadA, LB loadB, int K, v8f acc) {
  const int lane = threadIdx.x & 31;
  const int half = lane >> 4;   // 0 or 1
  const int r    = lane & 15;
  for (int k = 0; k < K; k += 4) {
    v2f a, b;
    a.x = loadA(r, k + 2*half);
    a.y = loadA(r, k + 2*half + 1);
    b.x = loadB(k + 2*half, r);
    b.y = loadB(k + 2*half + 1, r);
    acc = __builtin_amdgcn_wmma_f32_16x16x4_f32(
        /*neg_a=*/false, a, /*neg_b=*/false, b,
        /*c_mod=*/(short)0, acc, /*reuse_a=*/false, /*reuse_b=*/false);
  }
  return acc;
}

template <typename ST>
__device__ inline void wmma_store(v8f acc, ST store) {
  const int lane = threadIdx.x & 31;
  const int half = lane >> 4;
  const int r    = lane & 15;
#pragma unroll
  for (int i = 0; i < 8; ++i) store(i + 8*half, r, acc[i]);
}

// ---------------------------------------------------------------------------
// 1) s_i[row] = x_row . w_adj[0:128], s_j[row] = x_row . w_adj[128:256]
// ---------------------------------------------------------------------------
__global__ void k_scores(const float* __restrict__ x, const float* __restrict__ w_adj,
                         float* __restrict__ s_i, float* __restrict__ s_j) {
  __shared__ float pi[128], pj[128];
  const int row = blockIdx.x, c = threadIdx.x;
  const float xv = x[row*IN_DIM + c];
  pi[c] = xv * w_adj[c];
  pj[c] = xv * w_adj[IN_DIM + c];
  __syncthreads();
  for (int s = 64; s > 0; s >>= 1) {
    if (c < s) { pi[c] += pi[c+s]; pj[c] += pj[c+s]; }
    __syncthreads();
  }
  if (c == 0) { s_i[row] = pi[0]; s_j[row] = pj[0]; }
}

// ---------------------------------------------------------------------------
// 2) L2-normalize pseudo_labels rows into K=52 zero-padded buffer
// ---------------------------------------------------------------------------
__global__ void k_pnorm(const float* __restrict__ p, float* __restrict__ pn) {
  __shared__ float sm[64];
  const int row = blockIdx.x, c = threadIdx.x;
  const float v = (c < NCLS) ? p[row*NCLS + c] : 0.f;
  sm[c] = v * v;
  __syncthreads();
  for (int s = 32; s > 0; s >>= 1) { if (c < s) sm[c] += sm[c+s]; __syncthreads(); }
  float nrm = sqrtf(sm[0]);
  nrm = nrm < 1e-12f ? 1e-12f : nrm;
  const float inv = 1.f / nrm;
  if (c < NPAD) pn[row*NPAD + c] = (c < NCLS) ? v * inv : 0.f;
}

// ---------------------------------------------------------------------------
// 3) adj row softmax: logits = (s_i[t] + s_j[s] + b_adj) * scale
// ---------------------------------------------------------------------------
__global__ void k_adj(const float* __restrict__ s_i, const float* __restrict__ s_j,
                      const float* __restrict__ b_adj, float* __restrict__ adj) {
  __shared__ float red[256];
  const int row = blockIdx.x;            // 0..2047
  const int b   = row / TT;
  const int tid = threadIdx.x;           // 256 threads, 2 cols each
  const float si = s_i[row], ba = b_adj[0];
  const float l0 = (si + s_j[b*TT + tid]       + ba) * ADJ_SCALE;
  const float l1 = (si + s_j[b*TT + tid + 256] + ba) * ADJ_SCALE;
  red[tid] = fmaxf(l0, l1);
  __syncthreads();
  for (int s = 128; s > 0; s >>= 1) { if (tid < s) red[tid] = fmaxf(red[tid], red[tid+s]); __syncthreads(); }
  const float mx = red[0];
  __syncthreads();
  const float e0 = expf(l0 - mx), e1 = expf(l1 - mx);
  red[tid] = e0 + e1;
  __syncthreads();
  for (int s = 128; s > 0; s >>= 1) { if (tid < s) red[tid] += red[tid+s]; __syncthreads(); }
  const float inv = 1.f / red[0];
  adj[row*TT + tid]       = e0 * inv;
  adj[row*TT + tid + 256] = e1 * inv;
}

// ---------------------------------------------------------------------------
// 4) pdot[b] = (pn[b] @ pn[b]^T) / temp   (WMMA, K = 52)
// ---------------------------------------------------------------------------
__global__ void k_pdot(const float* __restrict__ pn, const float* __restrict__ pseudo_temp,
                       float* __restrict__ pdot) {
  const int wave = blockIdx.x * blockDim.y + threadIdx.y;   // 4096 waves
  const int b    = wave / 1024;
  const int rem  = wave % 1024;
  const int m0   = (rem / 32) * 16, n0 = (rem % 32) * 16;
  const float* P = pn + b*TT*NPAD;
  float t = fabsf(pseudo_temp[0]); t = t < 0.01f ? 0.01f : t;
  const float invT = 1.f / t;
  v8f acc = {};
  acc = wmma_loop([&](int r, int k) { return P[(m0+r)*NPAD + k]; },
                  [&](int k, int c) { return P[(n0+c)*NPAD + k]; },
                  NPAD, acc);
  wmma_store(acc, [&](int mi, int c, float v) {
    pdot[(b*TT + m0 + mi)*TT + n0 + c] = v * invT;
  });
}

// ---------------------------------------------------------------------------
// 5) per-row: softmax(pdot) vs adj, sum of squared diff * sigmoid(w)/(B*T*T)
// ---------------------------------------------------------------------------
__global__ void k_ploss(const float* __restrict__ pdot, const float* __restrict__ adj,
                        const float* __restrict__ sup_weight, float* __restrict__ rowsum) {
  __shared__ float red[256];
  const int row = blockIdx.x, tid = threadIdx.x;
  const float l0 = pdot[row*TT + tid], l1 = pdot[row*TT + tid + 256];
  red[tid] = fmaxf(l0, l1);
  __syncthreads();
  for (int s = 128; s > 0; s >>= 1) { if (tid < s) red[tid] = fmaxf(red[tid], red[tid+s]); __syncthreads(); }
  const float mx = red[0];
  __syncthreads();
  const float e0 = expf(l0 - mx), e1 = expf(l1 - mx);
  red[tid] = e0 + e1;
  __syncthreads();
  for (int s = 128; s > 0; s >>= 1) { if (tid < s) red[tid] += red[tid+s]; __syncthreads(); }
  const float inv = 1.f / red[0];
  __syncthreads();
  const float d0 = e0*inv - adj[row*TT + tid];
  const float d1 = e1*inv - adj[row*TT + tid + 256];
  red[tid] = d0*d0 + d1*d1;
  __syncthreads();
  for (int s = 128; s > 0; s >>= 1) { if (tid < s) red[tid] += red[tid+s]; __syncthreads(); }
  if (tid == 0) {
    const float sig = 1.f / (1.f + expf(-sup_weight[0]));
    rowsum[row] = red[0] * sig / (float)(BB * TT * TT);
  }
}

// 6) deterministic final reduce -> loss scalar
__global__ void k_reduce(const float* __restrict__ rowsum, float* __restrict__ loss) {
  __shared__ float red[256];
  const int tid = threadIdx.x;
  float s = 0.f;
  for (int i = tid; i < ROWS; i += 256) s += rowsum[i];
  red[tid] = s;
  __syncthreads();
  for (int k = 128; k > 0; k >>= 1) { if (tid < k) red[tid] += red[tid+k]; __syncthreads(); }
  if (tid == 0) loss[0] = red[0];
}

// ---------------------------------------------------------------------------
// 7) h1 = relu(x @ W1^T + b1)   M=2048 N=256 K=128   (WMMA)
// ---------------------------------------------------------------------------
__global__ void k_gemm1(const float* __restrict__ x, const float* __restrict__ W1,
                        const float* __restrict__ b1, float* __restrict__ h1) {
  const int wave = blockIdx.x * blockDim.y + threadIdx.y;   // 2048 waves
  const int m0 = (wave / (HID/16)) * 16, n0 = (wave % (HID/16)) * 16;
  v8f acc = {};
  acc = wmma_loop([&](int r, int k) { return x[(m0+r)*IN_DIM + k]; },
                  [&](int k, int c) { return W1[(n0+c)*IN_DIM + k]; },
                  IN_DIM, acc);
  wmma_store(acc, [&](int mi, int c, float v) {
    const float o = v + b1[n0 + c];
    h1[(m0+mi)*HID + n0 + c] = o > 0.f ? o : 0.f;
  });
}

// ---------------------------------------------------------------------------
// 8) h2[b] = adj[b] @ h1[b]   per-batch M=512 N=256 K=512   (WMMA)
// ---------------------------------------------------------------------------
__global__ void k_gemm2(const float* __restrict__ adj, const float* __restrict__ h1,
                        float* __restrict__ h2) {
  const int wave = blockIdx.x * blockDim.y + threadIdx.y;   // 2048 waves
  const int b = wave / 512, rem = wave % 512;
  const int m0 = (rem / 16) * 16, n0 = (rem % 16) * 16;
  const float* A  = adj + (size_t)b*TT*TT;
  const float* Bm = h1  + (size_t)b*TT*HID;
  v8f acc = {};
  acc = wmma_loop([&](int r, int k) { return A[(m0+r)*TT + k]; },
                  [&](int k, int c) { return Bm[k*HID + n0 + c]; },
                  TT, acc);
  wmma_store(acc, [&](int mi, int c, float v) {
    h2[((size_t)b*TT + m0 + mi)*HID + n0 + c] = v;
  });
}

// ---------------------------------------------------------------------------
// 9) y = h2 @ W2^T + b2 + x   M=2048 N=128 K=256   (WMMA)
// ---------------------------------------------------------------------------
__global__ void k_gemm3(const float* __restrict__ h2, const float* __restrict__ W2,
                        const float* __restrict__ b2, const float* __restrict__ x,
                        float* __restrict__ y) {
  const int wave = blockIdx.x * blockDim.y + threadIdx.y;   // 1024 waves
  const int m0 = (wave / (OUTD/16)) * 16, n0 = (wave % (OUTD/16)) * 16;
  v8f acc = {};
  acc = wmma_loop([&](int r, int k) { return h2[(m0+r)*HID + k]; },
                  [&](int k, int c) { return W2[(n0+c)*HID + k]; },
                  HID, acc);
  wmma_store(acc, [&](int mi, int c, float v) {
    const int row = m0 + mi, col = n0 + c;
    y[row*OUTD + col] = v + b2[col] + x[row*OUTD + col];
  });
}

// 10) LayerNorm rows of 128
__global__ void k_ln(const float* __restrict__ y, const float* __restrict__ gamma,
                     const float* __restrict__ beta, float* __restrict__ out) {
  __shared__ float red[128];
  const int row = blockIdx.x, c = threadIdx.x;
  const float v = y[row*OUTD + c];
  red[c] = v;
  __syncthreads();
  for (int s = 64; s > 0; s >>= 1) { if (c < s) red[c] += red[c+s]; __syncthreads(); }
  const float mu = red[0] * (1.f/OUTD);
  __syncthreads();
  const float d = v - mu;
  red[c] = d * d;
  __syncthreads();
  for (int s = 64; s > 0; s >>= 1) { if (c < s) red[c] += red[c+s]; __syncthreads(); }
  const float var = red[0] * (1.f/OUTD);
  out[row*OUTD + c] = d * rsqrtf(var + LN_EPS) * gamma[c] + beta[c];
}

// ---------------------------------------------------------------------------
extern "C" void kernel_launch(void* const* d_in, const int* in_sizes, int n_in,
                              void* d_out, int out_size, void* d_ws, size_t ws_size,
                              hipStream_t stream) {
  (void)in_sizes; (void)n_in; (void)out_size; (void)ws_size;
  const float* x     = (const float*)d_in[0];
  const float* pl    = (const float*)d_in[1];
  const float* W1    = (const float*)d_in[2];
  const float* b1    = (const float*)d_in[3];
  const float* W2    = (const float*)d_in[4];
  const float* b2    = (const float*)d_in[5];
  const float* w_adj = (const float*)d_in[6];
  const float* b_adj = (const float*)d_in[7];
  const float* ptemp = (const float*)d_in[8];
  const float* supw  = (const float*)d_in[9];
  const float* gamma = (const float*)d_in[10];
  const float* beta  = (const float*)d_in[11];

  float* out  = (float*)d_out;            // [2048,128]
  float* adj  = out + (size_t)ROWS*OUTD;  // [2048,512]
  float* loss = adj + (size_t)ROWS*TT;    // [1]

  // Workspace layout (floats). "big" region is reused: pdot first (loss path),
  // then h1|h2 (GNN path) — strictly ordered on the stream.
  float* ws     = (float*)d_ws;
  float* s_i    = ws;                       // 2048
  float* s_j    = s_i + ROWS;               // 2048
  float* pn     = s_j + ROWS;               // 2048*52
  float* rowsum = pn + (size_t)ROWS*NPAD;   // 2048
  float* big    = rowsum + ROWS;            // 1048576
  float* pdot   = big;                      // 4*512*512
  float* h1     = big;                      // 2048*256
  float* h2     = h1 + (size_t)ROWS*HID;    // 2048*256
  float* yb     = big + (size_t)ROWS*TT;    // 2048*128

  const dim3 wblk(32, 4);   // 4 waves / block, lane == threadIdx.x

  k_scores<<<ROWS, 128, 0, stream>>>(x, w_adj, s_i, s_j);
  k_pnorm <<<ROWS,  64, 0, stream>>>(pl, pn);
  k_adj   <<<ROWS, 256, 0, stream>>>(s_i, s_j, b_adj, adj);
  k_pdot  <<<1024, wblk, 0, stream>>>(pn, ptemp, pdot);
  k_ploss <<<ROWS, 256, 0, stream>>>(pdot, adj, supw, rowsum);
  k_reduce<<<1,    256, 0, stream>>>(rowsum, loss);
  k_gemm1 <<<512,  wblk, 0, stream>>>(x, W1, b1, h1);
  k_gemm2 <<<512,  wblk, 0, stream>>>(adj, h1, h2);
  k_gemm3 <<<256,  wblk, 0, stream>>>(h2, W2, b2, x, yb);
  k_ln    <<<ROWS, 128, 0, stream>>>(yb, gamma, beta, out);
}